// HybridAssigner_86998857548003
// MI455X (gfx1250) — compile-verified
//
#include <hip/hip_runtime.h>
#include <cstdint>
#include <cstddef>

// ---------------- problem constants (match reference) ----------------
#define NPTS   8192
#define NB     2
#define GX     334
#define GY     334
#define GZ     2
#define NVOX   (NB * GZ * GY * GX)          // 446224 voxels
#define WORDS  (NPTS / 32)                  // 256 u32 per adjacency row
#define D2MAX  0.36f                        // 0.6^2
#define PITER  16                           // propagate+jump iterations

typedef __attribute__((ext_vector_type(2))) float v2f;
typedef __attribute__((ext_vector_type(8))) float v8f;

// ---------------- ws layout (bytes) ----------------
// vox_cnt NVOX f32 | vox_sum NVOX*3 f32 | cxy 2N f32 | nrm N f32 |
// labelsA N i32 | labelsB N i32 | adjacency N*WORDS u32 (8 MB)
#define OFF_CNT   ((size_t)0)
#define OFF_SUM   (OFF_CNT + (size_t)NVOX * 4)
#define OFF_CXY   (OFF_SUM + (size_t)NVOX * 3 * 4)
#define OFF_NRM   (OFF_CXY + (size_t)NPTS * 2 * 4)
#define OFF_LABA  (OFF_NRM + (size_t)NPTS * 4)
#define OFF_LABB  (OFF_LABA + (size_t)NPTS * 4)
#define OFF_ADJ   (OFF_LABB + (size_t)NPTS * 4)

__device__ __forceinline__ int voxel_id(const float* p, int b) {
    int cx = (int)floorf((p[0] - (-50.0f)) / 0.3f);
    int cy = (int)floorf((p[1] - (-50.0f)) / 0.3f);
    int cz = (int)floorf((p[2] - (-5.0f)) / 6.0f);
    cx = min(max(cx, 0), GX - 1);
    cy = min(max(cy, 0), GY - 1);
    cz = min(max(cz, 0), GZ - 1);
    return ((b * GZ + cz) * GY + cy) * GX + cx;
}

// -------- 1. zero voxel accumulators --------
__global__ __launch_bounds__(256) void ha_zero(float* acc, int n) {
    int i = blockIdx.x * blockDim.x + threadIdx.x;
    if (i < n) acc[i] = 0.0f;
}

// -------- 2. scatter points into voxels --------
__global__ __launch_bounds__(256) void ha_scatter(const float* __restrict__ pts,
                                                  const int* __restrict__ bidx,
                                                  float* __restrict__ vcnt,
                                                  float* __restrict__ vsum) {
    int i = blockIdx.x * blockDim.x + threadIdx.x;
    if (i >= NPTS) return;
    const float p[3] = {pts[i*3+0], pts[i*3+1], pts[i*3+2]};
    int vid = voxel_id(p, bidx[i]);
    atomicAdd(&vcnt[vid], 1.0f);
    atomicAdd(&vsum[(size_t)vid*3+0], p[0]);
    atomicAdd(&vsum[(size_t)vid*3+1], p[1]);
    atomicAdd(&vsum[(size_t)vid*3+2], p[2]);
}

// -------- 3. gather per-point centers, write cpp/valid, build planes --------
__global__ __launch_bounds__(256) void ha_gather(const float* __restrict__ pts,
                                                 const int* __restrict__ bidx,
                                                 const float* __restrict__ vcnt,
                                                 const float* __restrict__ vsum,
                                                 v2f* __restrict__ cxy,
                                                 float* __restrict__ nrm,
                                                 float* __restrict__ out) {
    int i = blockIdx.x * blockDim.x + threadIdx.x;
    if (i >= NPTS) return;
    const float p[3] = {pts[i*3+0], pts[i*3+1], pts[i*3+2]};
    int vid = voxel_id(p, bidx[i]);
    float c = fmaxf(vcnt[vid], 1.0f);
    float mx = vsum[(size_t)vid*3+0] / c;
    float my = vsum[(size_t)vid*3+1] / c;
    float mz = vsum[(size_t)vid*3+2] / c;
    v2f xy; xy.x = mx; xy.y = my;
    cxy[i] = xy;
    nrm[i] = mx*mx + my*my;
    // outputs: cluster_inds [0,2N) | valid [2N,3N) | cpp [3N,6N)
    out[2*NPTS + i] = (vcnt[vid] >= 1.0f) ? 1.0f : 0.0f;
    out[3*NPTS + (size_t)i*3 + 0] = mx;
    out[3*NPTS + (size_t)i*3 + 1] = my;
    out[3*NPTS + (size_t)i*3 + 2] = mz;
}

// -------- 4. adjacency tiles via WMMA f32 16x16x4 --------
// One wave per 16-row x 128-col tile: 8 WMMAs, A operand + row-side scalars
// amortized across all sub-tiles; one full u32 bitmask word stored per 32 cols.
__global__ __launch_bounds__(32) void ha_adj(const v2f* __restrict__ cxy,
                                             const float* __restrict__ nrm,
                                             const int* __restrict__ bidx,
                                             unsigned* __restrict__ adj) {
    const int lane = threadIdx.x;            // 0..31, wave32
    const int i0 = blockIdx.x * 16;          // row tile base
    const int j0 = blockIdx.y * 128;         // col tile base (4 u32 words)
    const bool half = lane >= 16;            // lanes 16..31 hold K=2,3 (zeros)
    const int l15 = lane & 15;

    // A operand: 16x4, K = {x, y, 0, 0}  (ISA 16x4 f32 A layout)
    v2f arow = cxy[i0 + l15];
    v2f a; a.x = half ? 0.0f : arow.x; a.y = half ? 0.0f : arow.y;

    // row-side norms/batch for the 8 C/D rows this half-wave covers
    const int hv = half ? 8 : 0;
    float rn[8]; int rb[8];
    #pragma unroll
    for (int r = 0; r < 8; ++r) {
        rn[r] = nrm[i0 + hv + r];
        rb[r] = bidx[i0 + hv + r];
    }

    #pragma unroll
    for (int g = 0; g < 4; ++g) {            // 4 x 32-col word groups
        unsigned wb[16];
        #pragma unroll
        for (int r = 0; r < 16; ++r) wb[r] = 0u;

        #pragma unroll
        for (int s = 0; s < 2; ++s) {        // two 16-col WMMA sub-tiles
            int jc = j0 + g * 32 + s * 16 + l15;
            v2f bc = cxy[jc];
            v2f b; b.x = half ? 0.0f : bc.x; b.y = half ? 0.0f : bc.y;
            v8f c = {};
            v8f d = __builtin_amdgcn_wmma_f32_16x16x4_f32(
                false, a, false, b, (short)0, c, false, false);

            float cn = nrm[jc];
            int   cb = bidx[jc];
            if (g < 3 && s == 1) {           // prefetch next word group's cols
                __builtin_prefetch(&cxy[jc + 32], 0, 0);
                __builtin_prefetch(&nrm[jc + 32], 0, 0);
            }
            #pragma unroll
            for (int r = 0; r < 8; ++r) {
                int ir = i0 + hv + r;
                float d2 = rn[r] + cn - 2.0f * d[r];
                bool pred = ((d2 < D2MAX) && (rb[r] == cb)) || (ir == jc);
                unsigned m = __builtin_amdgcn_ballot_w32(pred);
                wb[r]     |= (m & 0xFFFFu) << (16 * s);  // lanes 0-15 -> row r
                wb[r + 8] |= (m >> 16)     << (16 * s);  // lanes 16-31 -> row r+8
            }
        }

        // wb[] is wave-uniform (ballot). Lane L<16 writes row i0+L's word.
        unsigned w = 0u;
        #pragma unroll
        for (int r = 0; r < 16; ++r) w = (lane == r) ? wb[r] : w;
        if (lane < 16) adj[(size_t)(i0 + lane) * WORDS + (j0 >> 5) + g] = w;
    }
}

// -------- 5. label init --------
__global__ __launch_bounds__(256) void ha_init_labels(int* lab) {
    int i = blockIdx.x * blockDim.x + threadIdx.x;
    if (i < NPTS) lab[i] = i;
}

// -------- 6. min-label propagation: one wave per row over the bitset --------
__global__ __launch_bounds__(32) void ha_propagate(const unsigned* __restrict__ adj,
                                                   const int* __restrict__ src,
                                                   int* __restrict__ dst) {
    const int row = blockIdx.x;
    const int lane = threadIdx.x;
    int m = 0x7FFFFFFF;
    const unsigned* rowp = adj + (size_t)row * WORDS;
    for (int w = lane; w < WORDS; w += 32) {
        unsigned bits = rowp[w];
        while (bits) {
            int b = __builtin_ctz(bits);
            bits &= bits - 1;
            int l = src[(w << 5) + b];
            m = min(m, l);
        }
    }
    #pragma unroll
    for (int off = 16; off > 0; off >>= 1) {
        int o = __shfl_xor(m, off, 32);
        m = min(m, o);
    }
    if (lane == 0) dst[row] = m;
}

// -------- 7. pointer jumping: dst[i] = src[src[i]] --------
__global__ __launch_bounds__(256) void ha_jump(const int* __restrict__ src,
                                               int* __restrict__ dst) {
    int i = blockIdx.x * blockDim.x + threadIdx.x;
    if (i < NPTS) dst[i] = src[src[i]];
}

// -------- 8. finalize outputs --------
__global__ __launch_bounds__(256) void ha_final(const int* __restrict__ bidx,
                                                const int* __restrict__ lab,
                                                float* __restrict__ out) {
    int i = blockIdx.x * blockDim.x + threadIdx.x;
    if (i >= NPTS) return;
    out[(size_t)i * 2 + 0] = (float)bidx[i];
    out[(size_t)i * 2 + 1] = (float)lab[i];   // every voxel holds >=1 point -> valid
}

extern "C" void kernel_launch(void* const* d_in, const int* in_sizes, int n_in,
                              void* d_out, int out_size, void* d_ws, size_t ws_size,
                              hipStream_t stream) {
    (void)in_sizes; (void)n_in; (void)out_size; (void)ws_size;
    const float* pts  = (const float*)d_in[0];
    const int*   bidx = (const int*)d_in[1];
    float* out = (float*)d_out;

    char* ws = (char*)d_ws;
    float*    vcnt = (float*)(ws + OFF_CNT);
    float*    vsum = (float*)(ws + OFF_SUM);
    v2f*      cxy  = (v2f*)(ws + OFF_CXY);
    float*    nrm  = (float*)(ws + OFF_NRM);
    int*      labA = (int*)(ws + OFF_LABA);
    int*      labB = (int*)(ws + OFF_LABB);
    unsigned* adj  = (unsigned*)(ws + OFF_ADJ);

    const int zn = NVOX * 4;                  // cnt + 3 sum planes are contiguous
    ha_zero<<<(zn + 255) / 256, 256, 0, stream>>>(vcnt, zn);
    ha_scatter<<<(NPTS + 255) / 256, 256, 0, stream>>>(pts, bidx, vcnt, vsum);
    ha_gather<<<(NPTS + 255) / 256, 256, 0, stream>>>(pts, bidx, vcnt, vsum,
                                                      cxy, nrm, out);

    dim3 agrid(NPTS / 16, NPTS / 128);
    ha_adj<<<agrid, 32, 0, stream>>>(cxy, nrm, bidx, adj);

    ha_init_labels<<<(NPTS + 255) / 256, 256, 0, stream>>>(labA);
    for (int it = 0; it < PITER; ++it) {
        ha_propagate<<<NPTS, 32, 0, stream>>>(adj, labA, labB);
        ha_jump<<<(NPTS + 255) / 256, 256, 0, stream>>>(labB, labA);
    }
    ha_final<<<(NPTS + 255) / 256, 256, 0, stream>>>(bidx, labA, out);
}